// CPCModule_28922309771359
// MI455X (gfx1250) — compile-verified
//
#include <hip/hip_runtime.h>
#include <hip/hip_bf16.h>
#include <math.h>

// CPC module for MI455X (gfx1250, wave32, WMMA).
// Sizes fixed by the reference: B=512, K=256, T=101 (p_sample=100), KH=128,
// G=3*KH=384, F=16, HW=256.

typedef __attribute__((ext_vector_type(2))) float v2f;
typedef __attribute__((ext_vector_type(8))) float v8f;

#define BB   512
#define KK   256
#define TT   101
#define KH   128
#define GG   384
#define FF   16
#define HW   256

// ---------------------------------------------------------------------------
// Kernel 1: gx[b, t, g] = sum_k seq[b,t,k] * W_ih[g,k]   (+ b_ih[g])
// seq[b,t,k] = z_q_x_st[b, k, t]  (t = flattened h*w index, contiguous dim)
// One wave computes one 16x16 C-tile via V_WMMA_F32_16X16X4_F32.
// grid = (G/16, ceil(T/16), B), block = 32 (one wave).
// ---------------------------------------------------------------------------
__global__ void k_gx(const float* __restrict__ zq, const float* __restrict__ Wih,
                     const float* __restrict__ bih, float* __restrict__ gx) {
    const int lane = threadIdx.x & 31;
    const int half = lane >> 4, l15 = lane & 15;
    const int g0 = blockIdx.x * 16;
    const int t0 = blockIdx.y * 16;
    const int b  = blockIdx.z;

    v8f acc = {};
    // A[i][kk] = zq[b, k0+kk, t0+i]  -> two b32 loads, stride 256 floats
    const float* ap = zq + (size_t)b * KK * HW + t0 + l15;
    // B[kk][j] = Wih[g0+j, k0+kk]    -> contiguous float2
    const float* bp = Wih + (size_t)(g0 + l15) * KK + half * 2;

    for (int k0 = 0; k0 < KK; k0 += 4) {
        const int kk = k0 + half * 2;
        v2f a; a.x = ap[(size_t)kk * HW]; a.y = ap[(size_t)(kk + 1) * HW];
        v2f bb = *(const v2f*)(bp + k0);
        acc = __builtin_amdgcn_wmma_f32_16x16x4_f32(false, a, false, bb,
                                                    (short)0, acc, false, false);
    }
    const float bias = bih[g0 + l15];
    #pragma unroll
    for (int e = 0; e < 8; ++e) {
        const int t = t0 + e + half * 8;
        if (t < TT)
            gx[((size_t)b * TT + t) * GG + g0 + l15] = acc[e] + bias;
    }
}

// ---------------------------------------------------------------------------
// Kernel 2 (launched 101x, stream-serialized): one GRU step.
// Fuses gh = h @ W_hh^T (+b_hh) for all three gates (r,z,n share the A tile),
// then the sigmoid/tanh gate math, producing h_new.
// grid = (KH/16, B/16), block = 32 (one wave).
// ---------------------------------------------------------------------------
__global__ void k_step(const float* __restrict__ gx, const float* __restrict__ Whh,
                       const float* __restrict__ bhh, const float* __restrict__ hin,
                       float* __restrict__ hout, int t) {
    const int lane = threadIdx.x & 31;
    const int half = lane >> 4, l15 = lane & 15;
    const int n0 = blockIdx.x * 16;   // hidden-column tile, 0..127
    const int m0 = blockIdx.y * 16;   // batch tile

    v8f accr = {}, accz = {}, accn = {};
    const float* ap  = hin + (size_t)(m0 + l15) * KH + half * 2;            // A = h tile
    const float* brp = Whh + (size_t)(      n0 + l15) * KH + half * 2;      // B gate r
    const float* bzp = Whh + (size_t)(KH  + n0 + l15) * KH + half * 2;      // B gate z
    const float* bnp = Whh + (size_t)(2*KH + n0 + l15) * KH + half * 2;     // B gate n

    for (int k0 = 0; k0 < KH; k0 += 4) {
        v2f a  = *(const v2f*)(ap  + k0);
        v2f br = *(const v2f*)(brp + k0);
        v2f bz = *(const v2f*)(bzp + k0);
        v2f bn = *(const v2f*)(bnp + k0);
        accr = __builtin_amdgcn_wmma_f32_16x16x4_f32(false, a, false, br, (short)0, accr, false, false);
        accz = __builtin_amdgcn_wmma_f32_16x16x4_f32(false, a, false, bz, (short)0, accz, false, false);
        accn = __builtin_amdgcn_wmma_f32_16x16x4_f32(false, a, false, bn, (short)0, accn, false, false);
    }
    const int   col   = n0 + l15;
    const float biasr = bhh[col], biasz = bhh[KH + col], biasn = bhh[2 * KH + col];

    #pragma unroll
    for (int e = 0; e < 8; ++e) {
        const int b = m0 + e + half * 8;
        const float* gxrow = gx + ((size_t)b * TT + t) * GG;
        const float xr = gxrow[col], xz = gxrow[KH + col], xn = gxrow[2 * KH + col];
        const float r = 1.f / (1.f + expf(-(xr + accr[e] + biasr)));
        const float z = 1.f / (1.f + expf(-(xz + accz[e] + biasz)));
        const float n = tanhf(xn + r * (accn[e] + biasn));
        const float hp = hin[(size_t)b * KH + col];
        hout[(size_t)b * KH + col] = (1.f - z) * n + z * hp;
    }
}

// ---------------------------------------------------------------------------
// Kernel 3: pred[f, b, k] = sum_h Wk_w[f,k,h] * c_t[b,h]  + Wk_b[f,k]
// grid = (K/16, B/16, F), block = 32.
// ---------------------------------------------------------------------------
__global__ void k_pred(const float* __restrict__ ct, const float* __restrict__ Wkw,
                       const float* __restrict__ Wkb, float* __restrict__ pred) {
    const int lane = threadIdx.x & 31;
    const int half = lane >> 4, l15 = lane & 15;
    const int n0 = blockIdx.x * 16;   // k-column tile
    const int m0 = blockIdx.y * 16;   // batch tile
    const int f  = blockIdx.z;

    v8f acc = {};
    const float* ap = ct  + (size_t)(m0 + l15) * KH + half * 2;
    const float* bp = Wkw + ((size_t)f * KK + n0 + l15) * KH + half * 2;
    for (int k0 = 0; k0 < KH; k0 += 4) {
        v2f a  = *(const v2f*)(ap + k0);
        v2f bb = *(const v2f*)(bp + k0);
        acc = __builtin_amdgcn_wmma_f32_16x16x4_f32(false, a, false, bb, (short)0, acc, false, false);
    }
    const float bias = Wkb[(size_t)f * KK + n0 + l15];
    #pragma unroll
    for (int e = 0; e < 8; ++e) {
        const int b = m0 + e + half * 8;
        pred[((size_t)f * BB + b) * KK + n0 + l15] = acc[e] + bias;
    }
}

// ---------------------------------------------------------------------------
// Kernel 4: totals[f, b, c] = sum_k enc[f,b,k] * pred[f,c,k]
// enc[f,b,k] = z_q_x_st[b, k, 101+f]
// grid = (B/16, B/16, F), block = 32.
// ---------------------------------------------------------------------------
__global__ void k_totals(const float* __restrict__ zq, const float* __restrict__ pred,
                         float* __restrict__ totals) {
    const int lane = threadIdx.x & 31;
    const int half = lane >> 4, l15 = lane & 15;
    const int n0 = blockIdx.x * 16;   // c tile
    const int m0 = blockIdx.y * 16;   // b tile
    const int f  = blockIdx.z;

    v8f acc = {};
    const float* ap = zq + (size_t)(m0 + l15) * KK * HW + (TT) + f;   // TT = 101 = p_sample+1
    const float* bp = pred + ((size_t)f * BB + n0 + l15) * KK + half * 2;
    for (int k0 = 0; k0 < KK; k0 += 4) {
        const int kk = k0 + half * 2;
        v2f a; a.x = ap[(size_t)kk * HW]; a.y = ap[(size_t)(kk + 1) * HW];
        v2f bb = *(const v2f*)(bp + k0);
        acc = __builtin_amdgcn_wmma_f32_16x16x4_f32(false, a, false, bb, (short)0, acc, false, false);
    }
    #pragma unroll
    for (int e = 0; e < 8; ++e) {
        const int b = m0 + e + half * 8;
        totals[((size_t)f * BB + b) * BB + n0 + l15] = acc[e];
    }
}

// ---------------------------------------------------------------------------
// Kernel 5: per-(f,b) logsumexp over c; dml[f,b] = totals[f,b,b] - lse.
// Also records lse for f==15 (needed for accuracy softmax).
// grid = F*B blocks, 256 threads.
// ---------------------------------------------------------------------------
__global__ void k_lse(const float* __restrict__ totals, float* __restrict__ dml,
                      float* __restrict__ lse15) {
    __shared__ float sm[256];
    const int fb = blockIdx.x, tid = threadIdx.x;
    const int f = fb >> 9, b = fb & (BB - 1);
    const float* row = totals + (size_t)fb * BB;

    float m = -3.402823466e38f;
    for (int c = tid; c < BB; c += 256) m = fmaxf(m, row[c]);
    sm[tid] = m; __syncthreads();
    for (int s = 128; s > 0; s >>= 1) {
        if (tid < s) sm[tid] = fmaxf(sm[tid], sm[tid + s]);
        __syncthreads();
    }
    const float M = sm[0]; __syncthreads();

    float sum = 0.f;
    for (int c = tid; c < BB; c += 256) sum += expf(row[c] - M);
    sm[tid] = sum; __syncthreads();
    for (int s = 128; s > 0; s >>= 1) {
        if (tid < s) sm[tid] += sm[tid + s];
        __syncthreads();
    }
    if (tid == 0) {
        const float lse = M + logf(sm[0]);
        dml[fb] = row[b] - lse;
        if (f == FF - 1) lse15[b] = lse;
    }
}

// ---------------------------------------------------------------------------
// Kernel 6: accuracy argmax. For each column c of totals[15]:
// argmax_b ( totals[15,b,c] - lse15[b] )  (softmax over c is monotone given lse).
// flags[c] = (argmax == c). grid = B blocks, 256 threads.
// ---------------------------------------------------------------------------
__global__ void k_acc(const float* __restrict__ t15, const float* __restrict__ lse15,
                      float* __restrict__ flags) {
    __shared__ float sv[256];
    __shared__ int   si[256];
    const int c = blockIdx.x, tid = threadIdx.x;

    float best = -3.402823466e38f; int bi = BB;
    for (int b = tid; b < BB; b += 256) {
        const float sc = t15[(size_t)b * BB + c] - lse15[b];
        if (sc > best) { best = sc; bi = b; }
    }
    sv[tid] = best; si[tid] = bi; __syncthreads();
    for (int s = 128; s > 0; s >>= 1) {
        if (tid < s) {
            if (sv[tid + s] > sv[tid] ||
                (sv[tid + s] == sv[tid] && si[tid + s] < si[tid])) {
                sv[tid] = sv[tid + s]; si[tid] = si[tid + s];
            }
        }
        __syncthreads();
    }
    if (tid == 0) flags[c] = (si[0] == c) ? 1.f : 0.f;
}

// ---------------------------------------------------------------------------
// Kernel 7: final scalars. out[0] = accuracy, out[1] = nce.
// ---------------------------------------------------------------------------
__global__ void k_final(const float* __restrict__ dml, const float* __restrict__ flags,
                        float* __restrict__ out) {
    __shared__ float sm[256];
    const int tid = threadIdx.x;

    float s = 0.f;
    for (int i = tid; i < FF * BB; i += 256) s += dml[i];
    sm[tid] = s; __syncthreads();
    for (int k = 128; k > 0; k >>= 1) {
        if (tid < k) sm[tid] += sm[tid + k];
        __syncthreads();
    }
    const float diagsum = sm[0]; __syncthreads();

    float c = 0.f;
    for (int i = tid; i < BB; i += 256) c += flags[i];
    sm[tid] = c; __syncthreads();
    for (int k = 128; k > 0; k >>= 1) {
        if (tid < k) sm[tid] += sm[tid + k];
        __syncthreads();
    }
    if (tid == 0) {
        out[1] = -diagsum / (float)(BB * FF);   // nce
        out[0] = sm[0] / (float)BB;             // accuracy
    }
}

// ---------------------------------------------------------------------------
extern "C" void kernel_launch(void* const* d_in, const int* in_sizes, int n_in,
                              void* d_out, int out_size, void* d_ws, size_t ws_size,
                              hipStream_t stream) {
    const float* zq_st  = (const float*)d_in[0];   // z_q_x_st (B,K,16,16)
    const float* zex    = (const float*)d_in[1];   // z_e_x    (passthrough)
    const float* zqx    = (const float*)d_in[2];   // z_q_x    (passthrough)
    const float* hidden = (const float*)d_in[3];   // (1,B,KH)
    const float* Wih    = (const float*)d_in[4];   // (384,256)
    const float* Whh    = (const float*)d_in[5];   // (384,128)
    const float* bih    = (const float*)d_in[6];
    const float* bhh    = (const float*)d_in[7];
    const float* Wkw    = (const float*)d_in[8];   // (16,256,128)
    const float* Wkb    = (const float*)d_in[9];   // (16,256)
    float* out = (float*)d_out;
    float* ws  = (float*)d_ws;

    // workspace layout (floats)
    size_t off = 0;
    float* gx     = ws + off; off += (size_t)BB * TT * GG;       // 19,857,408
    float* h0     = ws + off; off += (size_t)BB * KH;
    float* h1     = ws + off; off += (size_t)BB * KH;
    float* pred   = ws + off; off += (size_t)FF * BB * KK;
    float* totals = ws + off; off += (size_t)FF * BB * BB;
    float* lse15  = ws + off; off += BB;
    float* dml    = ws + off; off += (size_t)FF * BB;
    float* flags  = ws + off; off += BB;

    // h <- hidden[0]
    hipMemcpyAsync(h0, hidden, (size_t)BB * KH * sizeof(float),
                   hipMemcpyDeviceToDevice, stream);

    // gx = seq @ W_ih^T + b_ih
    k_gx<<<dim3(GG / 16, (TT + 15) / 16, BB), 32, 0, stream>>>(zq_st, Wih, bih, gx);

    // 101 sequential GRU steps (ping-pong h buffers)
    float* hin = h0; float* hout = h1;
    for (int t = 0; t < TT; ++t) {
        k_step<<<dim3(KH / 16, BB / 16), 32, 0, stream>>>(gx, Whh, bhh, hin, hout, t);
        float* tmp = hin; hin = hout; hout = tmp;
    }
    // final hidden state now in `hin`

    k_pred<<<dim3(KK / 16, BB / 16, FF), 32, 0, stream>>>(hin, Wkw, Wkb, pred);
    k_totals<<<dim3(BB / 16, BB / 16, FF), 32, 0, stream>>>(zq_st, pred, totals);
    k_lse<<<FF * BB, 256, 0, stream>>>(totals, dml, lse15);
    k_acc<<<BB, 256, 0, stream>>>(totals + (size_t)(FF - 1) * BB * BB, lse15, flags);
    k_final<<<1, 256, 0, stream>>>(dml, flags, out);

    // passthrough outputs
    const size_t npass = (size_t)BB * KK * HW;   // 33,554,432 floats each
    hipMemcpyAsync(out + 2,          zex, npass * sizeof(float),
                   hipMemcpyDeviceToDevice, stream);
    hipMemcpyAsync(out + 2 + npass,  zqx, npass * sizeof(float),
                   hipMemcpyDeviceToDevice, stream);
}